// DraftPicker_22333829939951
// MI455X (gfx1250) — compile-verified
//
#include <hip/hip_runtime.h>
#include <math.h>

typedef __attribute__((ext_vector_type(2))) float v2f;
typedef __attribute__((ext_vector_type(8))) float v8f;

#define POOL_N 45
#define PACK_N 15
#define EMB_E  512
#define HEAD_D 8
#define PFULL  47                       // POOL + win + rank
#define RSQRT_D 0.35355339059327373f    // 1/sqrt(8)

__global__ __launch_bounds__(128) void draft_picker_kernel(
    const int*   __restrict__ pool,   // [B,45]
    const int*   __restrict__ pack,   // [B,15]
    const float* __restrict__ wins,   // [B]
    const float* __restrict__ ranks,  // [B]
    const float* __restrict__ emb,    // [V,512]
    const float* __restrict__ win_w,  // [512]
    const float* __restrict__ win_b,  // [512]
    const float* __restrict__ rank_w, // [512]
    const float* __restrict__ rank_b, // [512]
    const float* __restrict__ q_w,    // [8,512]
    const float* __restrict__ k_w,    // [8,512]
    const float* __restrict__ v_w,    // [8,512]
    const float* __restrict__ sc_w,   // [8]
    const float* __restrict__ sc_b,   // [1]
    float*       __restrict__ out)    // [B,15]
{
    __shared__ __align__(16) float lds_zero[EMB_E];
    __shared__ __align__(16) float lds_aux[2][EMB_E];   // win row, rank row
    __shared__ __align__(16) float lds_q[16][HEAD_D];
    __shared__ __align__(16) float lds_k[48][HEAD_D];
    __shared__ __align__(16) float lds_v[48][HEAD_D];

    const int tid  = threadIdx.x;
    const int b    = blockIdx.x;
    const int wave = tid >> 5;          // 0..3
    const int lane = tid & 31;
    const int nrow = lane & 15;         // A row within tile / B column N
    const int koff = ((lane >> 4) & 1) * 2;  // K sub-offset for this lane half

    // ---- build synthetic rows (win/rank affine) and a zero row in LDS ----
    const float wv = wins[b];
    const float rv = ranks[b];
    for (int c = tid; c < EMB_E; c += 128) {
        lds_zero[c]   = 0.0f;
        lds_aux[0][c] = wv * win_w[c]  + win_b[c];
        lds_aux[1][c] = rv * rank_w[c] + rank_b[c];
    }
    __syncthreads();

    // ---- per-lane A-matrix row pointer (generic pointer: global or LDS) ----
    const float* aptr;
    if (wave == 0) {
        // pack tile (no relu); row 15 is zero padding
        aptr = (nrow < PACK_N) ? (emb + (size_t)pack[b * PACK_N + nrow] * EMB_E)
                               : (const float*)lds_zero;
    } else {
        const int r = (wave - 1) * 16 + nrow;      // pool_full row 0..47
        if (r < POOL_N)           aptr = emb + (size_t)pool[b * POOL_N + r] * EMB_E;
        else if (r == POOL_N)     aptr = lds_aux[0];
        else if (r == POOL_N + 1) aptr = lds_aux[1];
        else                      aptr = lds_zero; // row 47 padding
    }
    // Branchless relu: pool waves clamp at 0, pack wave clamps at -inf (no-op).
    const float lo = (wave == 0) ? -INFINITY : 0.0f;

    // ---- per-lane B-matrix row pointer: columns 0-7 = K proj, 8-15 = V proj ----
    const float* bptr;
    if (wave == 0) bptr = (nrow < 8) ? (q_w + nrow * EMB_E) : (const float*)lds_zero;
    else           bptr = (nrow < 8) ? (k_w + nrow * EMB_E) : (v_w + (nrow - 8) * EMB_E);

    // ---- main projection loop: D = A(16x4 f32) x B(4x16 f32) + C, fp32 ----
    v8f c0 = {};
    v8f c1 = {};
    #pragma unroll 4
    for (int k = 0; k < EMB_E; k += 8) {
        v2f a0 = *reinterpret_cast<const v2f*>(aptr + k + koff);
        v2f b0 = *reinterpret_cast<const v2f*>(bptr + k + koff);
        v2f a1 = *reinterpret_cast<const v2f*>(aptr + k + 4 + koff);
        v2f b1 = *reinterpret_cast<const v2f*>(bptr + k + 4 + koff);
        a0.x = fmaxf(a0.x, lo); a0.y = fmaxf(a0.y, lo);
        a1.x = fmaxf(a1.x, lo); a1.y = fmaxf(a1.y, lo);
        c0 = __builtin_amdgcn_wmma_f32_16x16x4_f32(false, a0, false, b0,
                                                   (short)0, c0, false, false);
        c1 = __builtin_amdgcn_wmma_f32_16x16x4_f32(false, a1, false, b1,
                                                   (short)0, c1, false, false);
    }
    v8f cacc = c0 + c1;

    // ---- scatter C tiles to LDS (C layout: lane L, vgpr j -> M=j+8*(L/16), N=L%16) ----
    const int rbase = ((lane >> 4) & 1) * 8;
    if (wave == 0) {
        if (nrow < 8) {
            #pragma unroll
            for (int j = 0; j < 8; ++j) lds_q[rbase + j][nrow] = cacc[j];
        }
    } else {
        const int gbase = (wave - 1) * 16 + rbase;
        if (nrow < 8) {
            #pragma unroll
            for (int j = 0; j < 8; ++j) lds_k[gbase + j][nrow] = cacc[j];
        } else {
            #pragma unroll
            for (int j = 0; j < 8; ++j) lds_v[gbase + j][nrow - 8] = cacc[j];
        }
    }
    __syncthreads();

    // ---- attention + scoring head: one thread per pack row ----
    if (tid < PACK_N) {
        float q[HEAD_D];
        #pragma unroll
        for (int d = 0; d < HEAD_D; ++d) q[d] = lds_q[tid][d];

        // pass 1: row max of relu(scores) (all >= 0)
        float m = 0.0f;
        for (int p = 0; p < PFULL; ++p) {
            float s = 0.0f;
            #pragma unroll
            for (int d = 0; d < HEAD_D; ++d) s += q[d] * lds_k[p][d];
            s = fmaxf(s * RSQRT_D, 0.0f);
            m = fmaxf(m, s);
        }
        // pass 2: exp-sum + context accumulation (recompute scores)
        float sum = 0.0f;
        float ctx[HEAD_D];
        #pragma unroll
        for (int d = 0; d < HEAD_D; ++d) ctx[d] = 0.0f;
        for (int p = 0; p < PFULL; ++p) {
            float s = 0.0f;
            #pragma unroll
            for (int d = 0; d < HEAD_D; ++d) s += q[d] * lds_k[p][d];
            s = fmaxf(s * RSQRT_D, 0.0f);
            float e = expf(s - m);
            sum += e;
            #pragma unroll
            for (int d = 0; d < HEAD_D; ++d) ctx[d] += e * lds_v[p][d];
        }
        const float inv = 1.0f / sum;
        float logit = sc_b[0];
        #pragma unroll
        for (int d = 0; d < HEAD_D; ++d) {
            float cd = ctx[d] * inv;
            cd = (cd > 0.0f) ? cd : 0.01f * cd;   // leaky relu
            logit += cd * sc_w[d];
        }
        out[b * PACK_N + tid] = logit;
    }
}

extern "C" void kernel_launch(void* const* d_in, const int* in_sizes, int n_in,
                              void* d_out, int out_size, void* d_ws, size_t ws_size,
                              hipStream_t stream) {
    const int*   pool   = (const int*)  d_in[0];
    const int*   pack   = (const int*)  d_in[1];
    const float* wins   = (const float*)d_in[2];
    const float* ranks  = (const float*)d_in[3];
    const float* emb    = (const float*)d_in[4];
    const float* win_w  = (const float*)d_in[5];
    const float* win_b  = (const float*)d_in[6];
    const float* rank_w = (const float*)d_in[7];
    const float* rank_b = (const float*)d_in[8];
    const float* q_w    = (const float*)d_in[9];
    const float* k_w    = (const float*)d_in[10];
    const float* v_w    = (const float*)d_in[11];
    const float* sc_w   = (const float*)d_in[12];
    const float* sc_b   = (const float*)d_in[13];
    float*       out    = (float*)d_out;

    const int nb = in_sizes[2];   // B from wins[]
    draft_picker_kernel<<<dim3(nb), dim3(128), 0, stream>>>(
        pool, pack, wins, ranks, emb, win_w, win_b, rank_w, rank_b,
        q_w, k_w, v_w, sc_w, sc_b, out);
}